// PhysicsAwareMSELoss_19310172962830
// MI455X (gfx1250) — compile-verified
//
#include <hip/hip_runtime.h>

#define ALPHA 0.7f
#define BETA  0.3f

#define AS1 __attribute__((address_space(1)))
#define AS3 __attribute__((address_space(3)))

typedef __attribute__((ext_vector_type(2))) float v2f;
typedef __attribute__((ext_vector_type(8))) float v8f;
typedef __attribute__((ext_vector_type(4))) int   v4i;

#ifndef __has_builtin
#define __has_builtin(x) 0
#endif

// ---------------------------------------------------------------------------
// CDNA5 async global->LDS copy (16B granule), with safe fallback.
// Builtin signature (confirmed by clang diagnostic):
//   (AS1 int4* global_src, AS3 int4* lds_dst, imm int offset, imm int cpol)
// ---------------------------------------------------------------------------
__device__ __forceinline__ void async_copy_b128(void* lds_dst, const float* gsrc) {
#if __has_builtin(__builtin_amdgcn_global_load_async_to_lds_b128)
    __builtin_amdgcn_global_load_async_to_lds_b128(
        (AS1 v4i*)const_cast<float*>(gsrc), (AS3 v4i*)lds_dst, 0, 0);
#else
    *reinterpret_cast<float4*>(lds_dst) = *reinterpret_cast<const float4*>(gsrc);
#endif
}

__device__ __forceinline__ void async_wait_all() {
#if __has_builtin(__builtin_amdgcn_s_wait_asynccnt)
    __builtin_amdgcn_s_wait_asynccnt(0);
#else
    asm volatile("s_wait_asynccnt 0x0" ::: "memory");
#endif
}

// Wait until at most 8 async ops outstanding: async loads complete in order,
// so after issuing 8 (chunk c) + 8 (chunk c+1), this guarantees chunk c is
// fully in LDS while chunk c+1 continues in flight.
__device__ __forceinline__ void async_wait_le8() {
#if __has_builtin(__builtin_amdgcn_s_wait_asynccnt)
    __builtin_amdgcn_s_wait_asynccnt(8);
#else
    asm volatile("s_wait_asynccnt 0x8" ::: "memory");
#endif
}

// ---------------------------------------------------------------------------
// Wave-level reduction of 4 independent per-lane floats via
// V_WMMA_F32_16X16X4_F32 (dot with a {0,1} selector matrix).
// A (16x4 fp32) over wave32: A.x carries K in {0,2} (value stream a),
// A.y carries K in {1,3} (value stream b). Selector B: B[0/2][n] = (n<8),
// B[1/3][n] = (n>=8)  =>  columns n<8 reduce stream a, n>=8 reduce stream b.
// Per-lane sum of the 8 D registers is a half-wave column sum; adding
// lane0+lane16 (resp. lane8+lane24) yields the full-wave sums.
// ---------------------------------------------------------------------------
__device__ __forceinline__ void wave_reduce4(float v0, float v1, float v2, float v3,
                                             float* s /* out[4] */) {
#if __has_builtin(__builtin_amdgcn_wmma_f32_16x16x4_f32)
    const int lane = threadIdx.x & 31;
    const float bx = ((lane & 15) < 8) ? 1.0f : 0.0f;
    v2f B; B.x = bx; B.y = 1.0f - bx;
    v8f c = {};
    v2f A; A.x = v0; A.y = v1;
    v8f d0 = __builtin_amdgcn_wmma_f32_16x16x4_f32(false, A, false, B,
                                                   (short)0, c, false, false);
    A.x = v2; A.y = v3;
    v8f d1 = __builtin_amdgcn_wmma_f32_16x16x4_f32(false, A, false, B,
                                                   (short)0, c, false, false);
    float r0 = 0.f, r1 = 0.f;
#pragma unroll
    for (int k = 0; k < 8; ++k) { r0 += d0[k]; r1 += d1[k]; }
    s[0] = __shfl(r0, 0, 32) + __shfl(r0, 16, 32);
    s[1] = __shfl(r0, 8, 32) + __shfl(r0, 24, 32);
    s[2] = __shfl(r1, 0, 32) + __shfl(r1, 16, 32);
    s[3] = __shfl(r1, 8, 32) + __shfl(r1, 24, 32);
#else
#pragma unroll
    for (int off = 16; off > 0; off >>= 1) {
        v0 += __shfl_xor(v0, off, 32);
        v1 += __shfl_xor(v1, off, 32);
        v2 += __shfl_xor(v2, off, 32);
        v3 += __shfl_xor(v3, off, 32);
    }
    s[0] = v0; s[1] = v1; s[2] = v2; s[3] = v3;
#endif
}

// ---------------------------------------------------------------------------
// ws accumulator layout (floats):
//  [0] sum_b mse_b * w_b          [1] sum elastic err (cnt>=2)
//  [2] count of samples cnt>=2    [3] sum mono partials
//  [4] sum D*tmask                [5] count tmask
//  [6] sum D^T*smask              [7] count smask
// ---------------------------------------------------------------------------
__global__ void pamse_init_kernel(float* __restrict__ acc) {
    if (threadIdx.x < 8) acc[threadIdx.x] = 0.0f;
}

// One block per sample; fully float4-vectorized single pass (S % (4*256)==0
// for this problem; boundary pair handled with one scalar load per vector).
__global__ void pamse_stats_kernel(const float* __restrict__ pred,
                                   const float* __restrict__ target,
                                   const float* __restrict__ strain,
                                   const int*   __restrict__ mat,
                                   const float* __restrict__ egpa,
                                   const float* __restrict__ mweights,
                                   float* __restrict__ acc, int S) {
    const int b   = blockIdx.x;
    const int tid = threadIdx.x;
    const float* p  = pred   + (size_t)b * S;
    const float* tg = target + (size_t)b * S;
    const float* ep = strain + (size_t)b * S;

    float msesum = 0.f, Sx = 0.f, Sy = 0.f, Sxy = 0.f, Sxx = 0.f, cnt = 0.f, mono = 0.f;
    const int stride = (int)blockDim.x * 4;
    for (int t4 = tid * 4; t4 + 3 < S; t4 += stride) {
        const float4 pv = *reinterpret_cast<const float4*>(p  + t4);
        const float4 tv = *reinterpret_cast<const float4*>(tg + t4);
        const float4 ev = *reinterpret_cast<const float4*>(ep + t4);

        float d;
        d = pv.x - tv.x; msesum += d * d;
        d = pv.y - tv.y; msesum += d * d;
        d = pv.z - tv.z; msesum += d * d;
        d = pv.w - tv.w; msesum += d * d;

        if (ev.x < 0.02f) { cnt += 1.f; Sx += ev.x; Sy += pv.x; Sxy += ev.x * pv.x; Sxx += ev.x * ev.x; }
        if (ev.y < 0.02f) { cnt += 1.f; Sx += ev.y; Sy += pv.y; Sxy += ev.y * pv.y; Sxx += ev.y * ev.y; }
        if (ev.z < 0.02f) { cnt += 1.f; Sx += ev.z; Sy += pv.z; Sxy += ev.z * pv.z; Sxx += ev.z * ev.z; }
        if (ev.w < 0.02f) { cnt += 1.f; Sx += ev.w; Sy += pv.w; Sxy += ev.w * pv.w; Sxx += ev.w * ev.w; }

        mono += fmaxf(pv.x - pv.y, 0.f);
        mono += fmaxf(pv.y - pv.z, 0.f);
        mono += fmaxf(pv.z - pv.w, 0.f);
        if (t4 + 4 < S) mono += fmaxf(pv.w - p[t4 + 4], 0.f);
    }

    __shared__ float red[8][8];
    float s[4];
    const int wave = tid >> 5;
    wave_reduce4(msesum, Sx, Sy, Sxy, s);
    if ((tid & 31) == 0) { red[wave][0]=s[0]; red[wave][1]=s[1]; red[wave][2]=s[2]; red[wave][3]=s[3]; }
    wave_reduce4(Sxx, cnt, mono, 0.f, s);
    if ((tid & 31) == 0) { red[wave][4]=s[0]; red[wave][5]=s[1]; red[wave][6]=s[2]; }
    __syncthreads();

    if (tid == 0) {
        float t0=0,t1=0,t2=0,t3=0,t4=0,t5=0,t6=0;
        const int nw = (int)(blockDim.x >> 5);
        for (int w = 0; w < nw; ++w) {
            t0+=red[w][0]; t1+=red[w][1]; t2+=red[w][2]; t3+=red[w][3];
            t4+=red[w][4]; t5+=red[w][5]; t6+=red[w][6];
        }
        float mse = t0 / (float)S;
        atomicAdd(&acc[0], mse * mweights[mat[b]]);

        float safe = fmaxf(t5, 1.0f);
        float cov  = t3 - t1 * t2 / safe;     // Sxy - Sx*Sy/cnt
        float var  = t4 - t1 * t1 / safe;     // Sxx - Sx^2/cnt
        float et   = egpa[b] * 1000.0f;
        float err  = fabsf(cov / (var + 1e-8f) - et) / (et + 1e-8f);
        if (t5 >= 2.0f) { atomicAdd(&acc[1], err); atomicAdd(&acc[2], 1.0f); }
        atomicAdd(&acc[3], t6);
    }
}

// ---------------------------------------------------------------------------
// Pairwise kernel: 16x16 pairs per block, stress rows double-buffered into
// LDS via CDNA5 async-to-LDS copies; each thread owns one (i,j) pair.
// Row padding of +4 floats makes the 16 j-lanes' ds_load_b128 reads hit
// disjoint bank groups (stride 260 floats -> bank offset 4*lj).
// ---------------------------------------------------------------------------
#define PK_CH  256
#define PK_PAD 4

__global__ void pamse_pair_kernel(const float* __restrict__ stress,
                                  const int*   __restrict__ mat,
                                  const float* __restrict__ temp,
                                  const float* __restrict__ sr,
                                  float* __restrict__ acc, int S) {
    __shared__ float sI[2][16][PK_CH + PK_PAD];
    __shared__ float sJ[2][16][PK_CH + PK_PAD];
    __shared__ float red[8][4];

    const int tid   = threadIdx.x;
    const int iBase = blockIdx.y * 16;
    const int jBase = blockIdx.x * 16;
    const int li = tid >> 4;
    const int lj = tid & 15;

    // Stage one chunk (32 rows x PK_CH floats = 2048 granules; 8 per thread).
    auto stage = [&](int buf, int c0) {
#pragma unroll
        for (int k = 0; k < 8; ++k) {
            int u   = k * 256 + tid;      // 0..2047
            int row = u >> 6;             // / (PK_CH/4)
            int c4  = u & 63;
            if (row < 16) {
                async_copy_b128(&sI[buf][row][c4 * 4],
                                stress + (size_t)(iBase + row) * S + c0 + c4 * 4);
            } else {
                int r = row - 16;
                async_copy_b128(&sJ[buf][r][c4 * 4],
                                stress + (size_t)(jBase + r) * S + c0 + c4 * 4);
            }
        }
    };

    const int nch = S / PK_CH;
    stage(0, 0);

    float a = 0.f;
    for (int c = 0; c < nch; ++c) {
        if (c + 1 < nch) {
            stage((c + 1) & 1, (c + 1) * PK_CH);   // overlap next chunk's DMA
            async_wait_le8();                      // chunk c complete (in-order)
        } else {
            async_wait_all();
        }
        __syncthreads();

        const float4* ri4 = reinterpret_cast<const float4*>(sI[c & 1][li]);
        const float4* rj4 = reinterpret_cast<const float4*>(sJ[c & 1][lj]);
#pragma unroll 4
        for (int t = 0; t < PK_CH / 4; ++t) {
            float4 x = ri4[t], y = rj4[t];
            a += fmaxf(x.x - y.x, 0.f);
            a += fmaxf(x.y - y.y, 0.f);
            a += fmaxf(x.z - y.z, 0.f);
            a += fmaxf(x.w - y.w, 0.f);
        }
        __syncthreads();
    }

    const int   i  = iBase + li, j = jBase + lj;
    const float D  = a / (float)S;
    const bool  same = (mat[i] == mat[j]);
    const float Ti = temp[i], Tj = temp[j];
    const float si = sr[i],   sj = sr[j];
    const float hi = fmaxf(si, sj), lo = fminf(si, sj);
    const float ratio = hi / (lo + 1e-8f);
    const float td  = Ti - Tj;
    const float atd = fabsf(td);
    // temp_loss mask at (i,j)
    const bool tm = same && (ratio <= 1.2f) && (atd >= 10.0f) && (td > 0.0f);
    // sr_loss uses D^T under smask: contribute D[i,j] * smask[j,i]
    const bool sm = same && (atd <= 10.0f) && (ratio >= 1.2f) && (sj > si);

    float s[4];
    wave_reduce4(tm ? D : 0.f, tm ? 1.f : 0.f, sm ? D : 0.f, sm ? 1.f : 0.f, s);
    const int wave = tid >> 5;
    if ((tid & 31) == 0) { red[wave][0]=s[0]; red[wave][1]=s[1]; red[wave][2]=s[2]; red[wave][3]=s[3]; }
    __syncthreads();
    if (tid == 0) {
        float t0=0,t1=0,t2=0,t3=0;
        for (int w = 0; w < 8; ++w) { t0+=red[w][0]; t1+=red[w][1]; t2+=red[w][2]; t3+=red[w][3]; }
        atomicAdd(&acc[4], t0); atomicAdd(&acc[5], t1);
        atomicAdd(&acc[6], t2); atomicAdd(&acc[7], t3);
    }
}

__global__ void pamse_final_kernel(const float* __restrict__ acc,
                                   float* __restrict__ out, int B, int S) {
    if (threadIdx.x == 0) {
        float mse     = acc[0] / (float)B;
        float elastic = acc[2] > 0.f ? acc[1] / acc[2] : 0.f;
        float mono    = acc[3] / ((float)B * (float)(S - 1));
        float tl      = acc[5] > 0.f ? acc[4] / acc[5] : 0.f;
        float sl      = acc[7] > 0.f ? acc[6] / acc[7] : 0.f;
        out[0] = ALPHA * mse + BETA * (elastic + mono + tl + sl);
    }
}

extern "C" void kernel_launch(void* const* d_in, const int* in_sizes, int n_in,
                              void* d_out, int out_size, void* d_ws, size_t ws_size,
                              hipStream_t stream) {
    const float* pred   = (const float*)d_in[0];
    const float* target = (const float*)d_in[1];
    const int*   mat    = (const int*)d_in[2];
    const float* strain = (const float*)d_in[3];
    const float* egpa   = (const float*)d_in[4];
    const float* temp   = (const float*)d_in[5];
    const float* sr     = (const float*)d_in[6];
    const float* mw     = (const float*)d_in[7];

    const int B = in_sizes[4];
    const int S = in_sizes[3] / B;
    float* acc = (float*)d_ws;

    pamse_init_kernel<<<1, 32, 0, stream>>>(acc);
    pamse_stats_kernel<<<dim3(B), dim3(256), 0, stream>>>(pred, target, strain,
                                                          mat, egpa, mw, acc, S);
    dim3 gridD(B / 16, B / 16);
    pamse_pair_kernel<<<gridD, dim3(256), 0, stream>>>(pred /* stress = pred[...,0] */,
                                                       mat, temp, sr, acc, S);
    pamse_final_kernel<<<1, 32, 0, stream>>>(acc, (float*)d_out, B, S);
}